// RankingLoss_67534065762412
// MI455X (gfx1250) — compile-verified
//
#include <hip/hip_runtime.h>
#include <hip/hip_bf16.h>
#include <stdint.h>

typedef __attribute__((ext_vector_type(2))) float v2f;
typedef __attribute__((ext_vector_type(8))) float v8f;

#define USE_ASYNC_LDS 1

namespace {
constexpr int   kS   = 512;          // sequence length
constexpr int   kB   = 256;          // batch
constexpr int   kTS  = 16;           // pair-tile edge (matches WMMA 16x16)
constexpr int   kBC  = 128;          // batch chunk staged in LDS
constexpr int   kRow = 2 * kBC + 4;  // dwords per LDS row (pad -> stride%64==4, conflict-free b128)
constexpr int   kTiles1D = kS / kTS;             // 32
constexpr int   kNumTiles = kTiles1D * kTiles1D; // 1024 (d_ws: 1024*2 floats = 8 KB)
constexpr float kMargin = 1.0f;
constexpr float kThresh = 0.01f;
constexpr float kEps    = 1e-8f;
}

// Exact-f32 batch reduction on the matrix pipe:
// D[m,n] = sum_{k<4} A[m,k] * 1 + C[m,n]  -> per-pair partial sums, column-replicated.
__device__ __forceinline__ v8f wmma_batchsum(v2f a, v8f c) {
  const v2f ones = {1.0f, 1.0f};
  return __builtin_amdgcn_wmma_f32_16x16x4_f32(
      /*neg_a=*/false, a, /*neg_b=*/false, ones,
      /*c_mod=*/(short)0, c, /*reuse_a=*/false, /*reuse_b=*/false);
}

#if USE_ASYNC_LDS
// CDNA5 async copy: global dword -> LDS at per-lane offset, per-lane scatter, tracked by ASYNCcnt.
__device__ __forceinline__ void async_cp_b32(float* ldsDst, const float* gSrc) {
  const unsigned la = (unsigned)(uintptr_t)ldsDst;  // low 32 bits of generic ptr == LDS byte offset
  asm volatile("global_load_async_to_lds_b32 %0, %1, off"
               :: "v"(la), "v"((unsigned long long)(uintptr_t)gSrc)
               : "memory");
}
__device__ __forceinline__ void wait_async0() {
  asm volatile("s_wait_asynccnt 0" ::: "memory");
}
#endif

__global__ __launch_bounds__(256)
void rankloss_tile(const float* __restrict__ pred,
                   const float* __restrict__ tgt,
                   float* __restrict__ part)
{
  const int jt  = blockIdx.x, it = blockIdx.y;
  const int bid = it * gridDim.x + jt;
  const int tid = threadIdx.x;

  if (jt < it) {  // tile entirely below the diagonal: contributes nothing
    if (tid == 0) { part[2 * bid] = 0.0f; part[2 * bid + 1] = 0.0f; }
    return;
  }

  __shared__ __align__(16) float lds[32 * kRow + 16];
  float* ldsI = lds;                 // 16 rows: columns i0..i0+15, (t,p) interleaved over batch
  float* ldsJ = lds + 16 * kRow;     // 16 rows: columns j0..j0+15
  float* wred = lds + 32 * kRow;     // 8 waves x (loss, count)

  const int i0 = it * kTS, j0 = jt * kTS;
  const int lane = tid & 31, w = tid >> 5;
  const int m = lane & 15, h = lane >> 4;   // A-matrix: M = lane%16, K-half = lane/16

  v8f zero = {0, 0, 0, 0, 0, 0, 0, 0};
  v8f Ce[2] = {zero, zero};   // per-pair err sums (rows = j-offset, cols replicated)
  v8f Cc[2] = {zero, zero};   // per-pair significance counts

  for (int bc = 0; bc < kB; bc += kBC) {
    __syncthreads();  // previous chunk's readers done before restaging
    // Coalesced global reads ([b][16 contiguous cols]) -> transposed LDS [col][b]
    for (int f = tid; f < kBC * kTS; f += 256) {
      const int  c  = f & (kTS - 1);
      const int  bl = f >> 4;
      const long ro = (long)(bc + bl) * kS;
      const int  off = c * kRow + bl * 2;
#if USE_ASYNC_LDS
      async_cp_b32(&ldsI[off],     tgt  + ro + i0 + c);
      async_cp_b32(&ldsI[off + 1], pred + ro + i0 + c);
      async_cp_b32(&ldsJ[off],     tgt  + ro + j0 + c);
      async_cp_b32(&ldsJ[off + 1], pred + ro + j0 + c);
#else
      ldsI[off]     = tgt [ro + i0 + c];
      ldsI[off + 1] = pred[ro + i0 + c];
      ldsJ[off]     = tgt [ro + j0 + c];
      ldsJ[off + 1] = pred[ro + j0 + c];
#endif
    }
#if USE_ASYNC_LDS
    wait_async0();
#endif
    __syncthreads();

    for (int ii = 0; ii < 2; ++ii) {          // each wave owns i-rows w and w+8
      const int  ir   = w + 8 * ii;
      const bool jgtI = (j0 + m) > (i0 + ir); // upper-triangle mask folded into sig
      const float* jrow = &ldsJ[m * kRow];
      const float* irow = &ldsI[ir * kRow];
      for (int st = 0; st < kBC / 4; ++st) {  // 4 batches per WMMA (K=4)
        const int o = st * 8 + h * 4;         // slot s -> batch st*4 + 2h + s
        const float4 jv = *(const float4*)(jrow + o);  // t_j[b], p_j[b], t_j[b+1], p_j[b+1]
        const float4 iv = *(const float4*)(irow + o);
        v2f aE, aC;
        {
          const float td = iv.x - jv.x, pd = iv.y - jv.y;
          const float at = fabsf(td);
          const bool  sg = jgtI && (at > kThresh);
          const float r  = __builtin_amdgcn_rcpf(at + kEps);   // v_rcp_f32 instead of IEEE div chain
          const float er = fmaxf(0.0f, kMargin - td * pd * r);
          aE.x = sg ? er : 0.0f;  aC.x = sg ? 1.0f : 0.0f;
        }
        {
          const float td = iv.z - jv.z, pd = iv.w - jv.w;
          const float at = fabsf(td);
          const bool  sg = jgtI && (at > kThresh);
          const float r  = __builtin_amdgcn_rcpf(at + kEps);
          const float er = fmaxf(0.0f, kMargin - td * pd * r);
          aE.y = sg ? er : 0.0f;  aC.y = sg ? 1.0f : 0.0f;
        }
        Ce[ii] = wmma_batchsum(aE, Ce[ii]);
        Cc[ii] = wmma_batchsum(aC, Cc[ii]);
      }
    }
  }

  // Finalize: rows of C are per-pair totals (replicated across columns).
  // Lane's 8 VGPRs hold rows c+8h; shfl_xor(16) adds the other half's rows.
  float L = 0.0f, C = 0.0f;
  for (int ii = 0; ii < 2; ++ii)
    for (int c = 0; c < 8; ++c) {
      const float cn = Cc[ii][c];
      if (cn > 0.5f) { L += Ce[ii][c] / cn; C += 1.0f; }
    }
  L += __shfl_xor(L, 16, 32);
  C += __shfl_xor(C, 16, 32);
  if (lane == 0) { wred[2 * w] = L; wred[2 * w + 1] = C; }
  __syncthreads();
  if (tid == 0) {
    float l = 0.0f, c = 0.0f;
    for (int k = 0; k < 8; ++k) { l += wred[2 * k]; c += wred[2 * k + 1]; }
    part[2 * bid] = l; part[2 * bid + 1] = c;   // deterministic: every slot rewritten each launch
  }
}

__global__ __launch_bounds__(256)
void rankloss_finalize(const float* __restrict__ part, float* __restrict__ out)
{
  __shared__ float sl[256], sc[256];
  const int t = threadIdx.x;
  float L = 0.0f, C = 0.0f;
  for (int k = t; k < kNumTiles; k += 256) { L += part[2 * k]; C += part[2 * k + 1]; }
  sl[t] = L; sc[t] = C;
  __syncthreads();
  for (int s = 128; s > 0; s >>= 1) {
    if (t < s) { sl[t] += sl[t + s]; sc[t] += sc[t + s]; }
    __syncthreads();
  }
  if (t == 0) out[0] = sl[0] / fmaxf(sc[0], 1.0f);
}

extern "C" void kernel_launch(void* const* d_in, const int* in_sizes, int n_in,
                              void* d_out, int out_size, void* d_ws, size_t ws_size,
                              hipStream_t stream)
{
  const float* pred = (const float*)d_in[0];   // predictions [256,512] f32
  const float* tgt  = (const float*)d_in[1];   // targets     [256,512] f32
  float* out  = (float*)d_out;                 // scalar f32
  float* part = (float*)d_ws;                  // 1024 tiles x (loss, count) = 8 KB
  (void)in_sizes; (void)n_in; (void)out_size; (void)ws_size;

  dim3 grid(kTiles1D, kTiles1D);
  rankloss_tile<<<grid, dim3(256), 0, stream>>>(pred, tgt, part);
  rankloss_finalize<<<dim3(1), dim3(256), 0, stream>>>(part, out);
}